// PairTabModel_12910671692115
// MI455X (gfx1250) — compile-verified
//
#include <hip/hip_runtime.h>
#include <hip/hip_bf16.h>
#include <stdint.h>

// Problem constants (from reference)
#define NFRAMES 8
#define NLOC    8192
#define NALL    10240
#define NNEI    256
#define NTYPES  4
#define NSPLINE 2048
// RMIN = 0.0f, HH = 0.02f

#define BLOCK_THREADS    1024
#define WAVES_PER_BLOCK  (BLOCK_THREADS / 32)      // 32 wave32s
#define LOCS_PER_WAVE    8
#define LOCS_PER_BLOCK   (WAVES_PER_BLOCK * LOCS_PER_WAVE)   // 256
#define BLOCKS_PER_FRAME (NLOC / LOCS_PER_BLOCK)             // 32

// Dynamic LDS layout:
//   [0, NALL*16)                   float4 per atom: {x,y,z, atype bits}
//   [NALL*16, +WAVES*2*NNEI*4)     per-wave double-buffered nlist rows
#define LDS_COORD_BYTES  (NALL * 16)                      // 163840
#define LDS_NLIST_BYTES  (WAVES_PER_BLOCK * 2 * NNEI * 4) // 65536
#define LDS_TOTAL_BYTES  (LDS_COORD_BYTES + LDS_NLIST_BYTES) // 229376 < 320KB/WGP

// Async copy of one nlist row (256 ints = 1KB) into LDS: each of the 32 lanes
// moves 32 bytes as two b128 async-to-LDS transfers (CDNA5, ASYNCcnt-tracked).
// nlist is a read-once 64MiB stream -> non-temporal hint keeps it out of the
// hot L2 working set (coord + tab tables).
__device__ __forceinline__ void async_fill_nlist_row(const int* gsrc,
                                                     unsigned lds_byte_off,
                                                     int lane) {
  unsigned long long g = (unsigned long long)(uintptr_t)(gsrc + lane * 8);
  unsigned l = lds_byte_off + (unsigned)lane * 32u;
  asm volatile("global_load_async_to_lds_b128 %0, %1, off th:TH_LOAD_NT"
               :: "v"(l), "v"(g) : "memory");
  asm volatile("global_load_async_to_lds_b128 %0, %1, off offset:16 th:TH_LOAD_NT"
               :: "v"(l), "v"(g) : "memory");
}

extern "C" __global__ __launch_bounds__(BLOCK_THREADS)
void pairtab_energy_kernel(const float* __restrict__ coord,   // (NF,NALL,3)
                           const int*   __restrict__ atype,   // (NF,NALL)
                           const int*   __restrict__ nlist,   // (NF,NLOC,NNEI)
                           const float* __restrict__ tab,     // (NT,NT,NSPLINE,4)
                           float*       __restrict__ out) {   // (NF,NLOC)
  extern __shared__ __align__(16) unsigned char lds_raw[];
  float4* sh4  = (float4*)lds_raw;
  int*    nbuf = (int*)(lds_raw + LDS_COORD_BYTES);

  const int f    = blockIdx.y;
  const int tid  = threadIdx.x;
  const int wave = tid >> 5;
  const int lane = tid & 31;

  const int  loc0   = blockIdx.x * LOCS_PER_BLOCK + wave * LOCS_PER_WAVE;
  int*       wbuf   = nbuf + wave * (2 * NNEI);
  const unsigned wb_off = (unsigned)(uintptr_t)wbuf;  // low 32 bits = LDS offset
  const int* nrow   = nlist + ((long)f * NLOC + loc0) * (long)NNEI;

  // Prime the async nlist pipeline for loc0 -> buffer 0 BEFORE staging, so its
  // HBM latency hides under the coord-table staging below (disjoint LDS region,
  // per-wave private buffer, ASYNCcnt-tracked -> no barrier interaction).
  async_fill_nlist_row(nrow, wb_off, lane);

  // ---- Stage the whole frame's coord+type table into LDS (coalesced) ----
  {
    const float* cF = coord + (size_t)f * NALL * 3;
    const int*   aF = atype + (size_t)f * NALL;
    for (int a = tid; a < NALL; a += BLOCK_THREADS) {
      float4 v;
      v.x = cF[a * 3 + 0];
      v.y = cF[a * 3 + 1];
      v.z = cF[a * 3 + 2];
      v.w = __int_as_float(aF[a]);
      sh4[a] = v;   // ds_store_b128
    }
  }
  __syncthreads();

  const float4* tab4 = (const float4*)tab;

  for (int i = 0; i < LOCS_PER_WAVE; ++i) {
    // Issue next row's async copy, then wait only for the row we need now.
    if (i + 1 < LOCS_PER_WAVE) {
      async_fill_nlist_row(nrow + (i + 1) * NNEI,
                           wb_off + ((i & 1) ? 0u : (unsigned)(NNEI * 4)),
                           lane);
      asm volatile("s_wait_asynccnt 0x2" ::: "memory");  // oldest fill done
    } else {
      asm volatile("s_wait_asynccnt 0x0" ::: "memory");
    }

    const int    loc = loc0 + i;
    const float4 ci  = sh4[loc];                 // central atom (loc < NLOC <= NALL)
    const int    it  = __float_as_int(ci.w);     // i_type
    const int*   cur = wbuf + (i & 1) * NNEI;

    const int4* lp = (const int4*)(cur + lane * 8);
    const int4 ja = lp[0];
    const int4 jb = lp[1];
    const int jj[8] = {ja.x, ja.y, ja.z, ja.w, jb.x, jb.y, jb.z, jb.w};

    float acc = 0.0f;
#pragma unroll
    for (int k = 0; k < 8; ++k) {
      const int  j     = jj[k];
      const bool valid = (j >= 0);
      const int  jc    = valid ? j : 0;            // masked_nlist = clip(nlist, 0)

      const float4 ca = sh4[jc];                   // ds_load_b128 gather from LDS
      const float dx = ci.x - ca.x;
      const float dy = ci.y - ca.y;
      const float dz = ci.z - ca.z;
      const float rr = sqrtf(dx * dx + dy * dy + dz * dz);

      const float uu  = (rr - 0.0f) / 0.02f;       // (rr - RMIN) / HH, as reference
      const int   idx = (int)uu;                   // trunc == floor for uu >= 0
      const float fu  = uu - (float)idx;
      int c = idx;
      c = c < 0 ? 0 : c;
      c = c > (NSPLINE - 1) ? (NSPLINE - 1) : c;

      const int jt = __float_as_int(ca.w);         // j_type
      const float4 cf = tab4[(size_t)(it * NTYPES + jt) * NSPLINE + c]; // L2-hot

      const float e = ((cf.x * fu + cf.y) * fu + cf.z) * fu + cf.w;
      acc += (valid && (idx < NSPLINE)) ? e : 0.0f;
    }

    // Wave32 reduction over the 32 lanes' partial sums
#pragma unroll
    for (int off = 16; off; off >>= 1)
      acc += __shfl_xor(acc, off, 32);

    if (lane == 0)
      out[(size_t)f * NLOC + loc] = 0.5f * acc;
  }
}

extern "C" void kernel_launch(void* const* d_in, const int* in_sizes, int n_in,
                              void* d_out, int out_size, void* d_ws, size_t ws_size,
                              hipStream_t stream) {
  (void)in_sizes; (void)n_in; (void)out_size; (void)d_ws; (void)ws_size;

  const float* coord = (const float*)d_in[0];  // extended_coord  f32
  const int*   atype = (const int*)  d_in[1];  // extended_atype  i32
  const int*   nlist = (const int*)  d_in[2];  // nlist           i32
  const float* tab   = (const float*)d_in[3];  // tab_data        f32
  float*       out   = (float*)d_out;          // (NFRAMES, NLOC) f32

  // Opt in to >64KB dynamic LDS (CDNA5 WGP has 320KB). Deterministic, capture-safe.
  (void)hipFuncSetAttribute((const void*)pairtab_energy_kernel,
                            hipFuncAttributeMaxDynamicSharedMemorySize,
                            LDS_TOTAL_BYTES);

  dim3 grid(BLOCKS_PER_FRAME, NFRAMES);
  pairtab_energy_kernel<<<grid, BLOCK_THREADS, LDS_TOTAL_BYTES, stream>>>(
      coord, atype, nlist, tab, out);
}